// DTC_49503793054575
// MI455X (gfx1250) — compile-verified
//
#include <hip/hip_runtime.h>
#include <hip/hip_bf16.h>

// CDNA5 / gfx1250. wave32. Uses V_WMMA_F32_16X16X4_F32 for the fp32 GEMMs.

typedef __attribute__((ext_vector_type(2))) float v2f;
typedef __attribute__((ext_vector_type(8))) float v8f;

#define CIN  256
#define COUT 256
#define HLO  96
#define NLO  (HLO*HLO)   // 9216 low-res pixels per (b, channel)
#define NB   4
#define HS   192         // upsampled H=W

// ---------------------------------------------------------------------------
// Kernel 1: fused GEMMs at low resolution.
//   y_base[b,o,hw] = sum_c w_base[o,c]    * x[b,c,hw]      (M=256,K=256,N=9216)
//   y_proc[b,o,hw] = sum_c w_process[o,c] * x[b,c,hw]      (M=256,K=256,N=9216)
//   coord [b,q,hw] = sum_c w_tc[c*16+q]   * x[b,c,hw]      (M=16, K=256,N=9216)
// One workgroup (8 waves) per (ntile of 16 columns, batch).  B-tile (256x16
// fp32, 16KB) staged in LDS once and reused by all 33 M-tile tasks.
// ---------------------------------------------------------------------------
__global__ __launch_bounds__(256) void DTC_gemm_kernel(
    const float* __restrict__ x,         // [B,256,96,96]
    const float* __restrict__ w_process, // [256,256]
    const float* __restrict__ w_base,    // [256,256]
    const float* __restrict__ w_tc,      // [256,4,2,2] viewed as [256][16]
    float* __restrict__ y_base,          // [B,256,9216]
    float* __restrict__ y_proc,          // [B,256,9216]
    float* __restrict__ coord)           // [B,16,9216]
{
    __shared__ float Bs[CIN * 16];       // [K=256][N=16]

    const int ntile = blockIdx.x;        // 0..575
    const int b     = blockIdx.y;        // 0..3
    const int hw0   = ntile * 16;
    const int t     = threadIdx.x;

    // ---- stage B tile: each thread copies float4 chunks, full 64B lines ----
    {
        const float* xb = x + (size_t)b * CIN * NLO + hw0;
        const int c0  = t >> 2;          // 0..63
        const int seg = (t & 3) * 4;     // 0,4,8,12
        #pragma unroll
        for (int it = 0; it < 4; ++it) {
            const int cc = c0 + it * 64;
            const float4 v = *(const float4*)(xb + (size_t)cc * NLO + seg);
            *(float4*)(&Bs[cc * 16 + seg]) = v;
        }
    }
    __syncthreads();

    const int lane  = t & 31;
    const int wave  = t >> 5;
    const int n     = lane & 15;         // column within tile / M row for A
    const int khalf = lane >> 4;         // 0: K pair {0,1}; 1: K pair {2,3}

    // 33 tasks: 0..15 -> w_base M-tiles, 16..31 -> w_process, 32 -> coord
    for (int task = wave; task < 33; task += 8) {
        v8f acc = {};
        if (task < 32) {
            const float* W    = (task < 16) ? w_base : w_process;
            const int  mbase  = (task & 15) * 16;
            const int  mrow   = mbase + n;  // A layout: lane&15 == M row
            const float2* arow = (const float2*)(W + (size_t)mrow * CIN);
            #pragma unroll 4
            for (int k0 = 0; k0 < CIN; k0 += 4) {
                const float2 af = arow[(k0 >> 1) + khalf]; // W[m][k0+2*khalf ..+1]
                v2f a;  a[0] = af.x;  a[1] = af.y;
                const int krow = k0 + 2 * khalf;
                v2f bf; bf[0] = Bs[krow * 16 + n]; bf[1] = Bs[(krow + 1) * 16 + n];
                acc = __builtin_amdgcn_wmma_f32_16x16x4_f32(
                        false, a, false, bf, (short)0, acc, false, false);
            }
            float* out = (task < 16) ? y_base : y_proc;
            #pragma unroll
            for (int r = 0; r < 8; ++r) {
                const int m = mbase + r + (khalf ? 8 : 0); // C/D layout
                out[((size_t)b * COUT + m) * NLO + hw0 + n] = acc[r];
            }
        } else {
            // coord GEMM: A[q][c] = w_tc[c*16 + q] (transposed read, coalesced in q)
            #pragma unroll 4
            for (int k0 = 0; k0 < CIN; k0 += 4) {
                const int kk = k0 + 2 * khalf;
                v2f a;  a[0] = w_tc[kk * 16 + n];       a[1] = w_tc[(kk + 1) * 16 + n];
                v2f bf; bf[0] = Bs[kk * 16 + n];        bf[1] = Bs[(kk + 1) * 16 + n];
                acc = __builtin_amdgcn_wmma_f32_16x16x4_f32(
                        false, a, false, bf, (short)0, acc, false, false);
            }
            #pragma unroll
            for (int r = 0; r < 8; ++r) {
                const int q = r + (khalf ? 8 : 0);
                coord[((size_t)b * 16 + q) * NLO + hw0 + n] = acc[r];
            }
        }
    }
}

// ---------------------------------------------------------------------------
// Kernel 2: per output pixel, fold (grid-sample bilinear) x (upsample
// bilinear) into 16 channel-independent low-res taps on y_proc, plus 4
// upsample taps on y_base; loop 256 channels with coalesced stores.
// ---------------------------------------------------------------------------
__device__ __forceinline__ void up_taps(int u, int& i0, int& i1, float& w1) {
    // align_corners=False, scale 2, n=96: src = max((u+0.5)/2 - 0.5, 0)
    const float src = fmaxf((u + 0.5f) * 0.5f - 0.5f, 0.0f);
    const float f   = floorf(src);
    i0 = (int)f;
    i1 = min(i0 + 1, HLO - 1);
    w1 = src - f;
}

__global__ __launch_bounds__(256) void DTC_sample_kernel(
    const float* __restrict__ y_base,
    const float* __restrict__ y_proc,
    const float* __restrict__ coord,
    float* __restrict__ out)             // [B,256,192,192]
{
    const int gid = blockIdx.x * 256 + threadIdx.x;
    if (gid >= NB * HS * HS) return;
    const int ws = gid % HS;
    const int hs = (gid / HS) % HS;
    const int b  = gid / (HS * HS);

    // coord channel: q = o*4 + (hs&1)*2 + (ws&1), at low-res pixel (i,j)
    const int i  = hs >> 1, j = ws >> 1;
    const int q0 = ((hs & 1) << 1) | (ws & 1);
    const float* cb = coord + (size_t)b * 16 * NLO + i * HLO + j;
    const float cx  = cb[(size_t)(0 + q0)  * NLO];
    const float cy  = cb[(size_t)(4 + q0)  * NLO];
    const float cwx = cb[(size_t)(8 + q0)  * NLO];
    const float cwy = cb[(size_t)(12 + q0) * NLO];

    const float lam = 1.0f / 96.0f;
    const float dx = lam * tanhf(cx) * (1.0f / (1.0f + __expf(-cwx)));
    const float dy = lam * tanhf(cy) * (1.0f / (1.0f + __expf(-cwy)));
    const float gx = -1.0f + 2.0f * ws / (float)(HS - 1);
    const float gy = -1.0f + 2.0f * hs / (float)(HS - 1);
    // grid_sample on 192x192 image, align_corners=False
    const float ix = (gx + dx + 1.0f) * (HS * 0.5f) - 0.5f;
    const float iy = (gy + dy + 1.0f) * (HS * 0.5f) - 0.5f;
    const float fx = floorf(ix), fy = floorf(iy);
    const int   x0 = (int)fx,    y0 = (int)fy;
    const float wx = ix - fx,    wy = iy - fy;

    int   idxD[16];
    float wD[16];
    #pragma unroll
    for (int nb = 0; nb < 4; ++nb) {
        const int X = x0 + (nb & 1);
        const int Y = y0 + (nb >> 1);
        const float wn = ((nb & 1) ? wx : 1.0f - wx) * ((nb >> 1) ? wy : 1.0f - wy);
        const bool valid = (X >= 0) && (X < HS) && (Y >= 0) && (Y < HS);
        const int Xc = valid ? X : 0;
        const int Yc = valid ? Y : 0;
        const float wv = valid ? wn : 0.0f;
        int jx0, jx1, jy0, jy1; float ux, uy;
        up_taps(Xc, jx0, jx1, ux);
        up_taps(Yc, jy0, jy1, uy);
        idxD[nb * 4 + 0] = jy0 * HLO + jx0; wD[nb * 4 + 0] = wv * (1.0f - uy) * (1.0f - ux);
        idxD[nb * 4 + 1] = jy0 * HLO + jx1; wD[nb * 4 + 1] = wv * (1.0f - uy) * ux;
        idxD[nb * 4 + 2] = jy1 * HLO + jx0; wD[nb * 4 + 2] = wv * uy * (1.0f - ux);
        idxD[nb * 4 + 3] = jy1 * HLO + jx1; wD[nb * 4 + 3] = wv * uy * ux;
    }

    int bx0, bx1, by0, by1; float ax, ay;
    up_taps(ws, bx0, bx1, ax);
    up_taps(hs, by0, by1, ay);
    const int   idxB[4] = { by0 * HLO + bx0, by0 * HLO + bx1,
                            by1 * HLO + bx0, by1 * HLO + bx1 };
    const float wB[4]   = { (1.0f - ay) * (1.0f - ax), (1.0f - ay) * ax,
                            ay * (1.0f - ax),          ay * ax };

    const float* pb = y_base + (size_t)b * COUT * NLO;
    const float* pp = y_proc + (size_t)b * COUT * NLO;
    float* po = out + (size_t)b * COUT * HS * HS + (size_t)hs * HS + ws;

    for (int c = 0; c < COUT; ++c) {
        const float* rb = pb + (size_t)c * NLO;
        const float* rp = pp + (size_t)c * NLO;
        float acc = wB[0] * rb[idxB[0]] + wB[1] * rb[idxB[1]]
                  + wB[2] * rb[idxB[2]] + wB[3] * rb[idxB[3]];
        #pragma unroll
        for (int tp = 0; tp < 16; ++tp)
            acc = fmaf(wD[tp], rp[idxD[tp]], acc);
        po[(size_t)c * HS * HS] = acc;
    }
}

extern "C" void kernel_launch(void* const* d_in, const int* in_sizes, int n_in,
                              void* d_out, int out_size, void* d_ws, size_t ws_size,
                              hipStream_t stream) {
    const float* x         = (const float*)d_in[0];  // [4,256,96,96]
    const float* w_process = (const float*)d_in[1];  // [256,256]
    const float* w_base    = (const float*)d_in[2];  // [256,256]
    const float* w_tc      = (const float*)d_in[3];  // [256,4,2,2]
    float* out = (float*)d_out;                      // [4,256,192,192]

    float* ws     = (float*)d_ws;
    float* y_base = ws;                                  // 4*256*9216 floats
    float* y_proc = y_base + (size_t)NB * COUT * NLO;    // 4*256*9216 floats
    float* coord  = y_proc + (size_t)NB * COUT * NLO;    // 4*16*9216  floats

    dim3 g1(NLO / 16, NB);   // 576 x 4 workgroups
    DTC_gemm_kernel<<<g1, 256, 0, stream>>>(x, w_process, w_base, w_tc,
                                            y_base, y_proc, coord);

    const int total = NB * HS * HS;  // 147456 output pixels
    DTC_sample_kernel<<<(total + 255) / 256, 256, 0, stream>>>(y_base, y_proc,
                                                               coord, out);
}